// GIN_77446850281718
// MI455X (gfx1250) — compile-verified
//
#include <hip/hip_runtime.h>
#include <hip/hip_bf16.h>

typedef __attribute__((ext_vector_type(2))) float v2f;
typedef __attribute__((ext_vector_type(8))) float v8f;

#define HID 64
#define LAYERS 4

// ---------------------------------------------------------------------------
// CDNA5 async global<->LDS copies (ASYNCcnt). Per-lane LDS addresses let us
// keep padded / transposed LDS layouts while the DMA does the data movement.
// ---------------------------------------------------------------------------
__device__ __forceinline__ void gin_async_ld_b128(unsigned lds_addr,
                                                  const void* gaddr) {
    asm volatile("global_load_async_to_lds_b128 %0, %1, off"
                 :: "v"(lds_addr), "v"(gaddr)
                 : "memory");
}

__device__ __forceinline__ void gin_async_ld_b32(unsigned lds_addr,
                                                 const void* gaddr) {
    asm volatile("global_load_async_to_lds_b32 %0, %1, off"
                 :: "v"(lds_addr), "v"(gaddr)
                 : "memory");
}

__device__ __forceinline__ void gin_async_st_b128(const void* gaddr,
                                                  unsigned lds_addr) {
    asm volatile("global_store_async_from_lds_b128 %0, %1, off"
                 :: "v"(gaddr), "v"(lds_addr)
                 : "memory");
}

__device__ __forceinline__ void gin_async_wait0() {
    asm volatile("s_wait_asynccnt 0x0" ::: "memory");
}

// ---------------------------------------------------------------------------
// Generic f32 zero-fill
// ---------------------------------------------------------------------------
__global__ void gin_zero_f32(float* __restrict__ p, int n) {
    int i = blockIdx.x * blockDim.x + threadIdx.x;
    if (i < n) p[i] = 0.0f;
}

// ---------------------------------------------------------------------------
// Scatter-add neighbor aggregation: agg[dst] += x[src]
// 16 lanes per edge, float4 per lane (64 channels).
// ---------------------------------------------------------------------------
__global__ void gin_scatter_add(const float* __restrict__ x,
                                const int* __restrict__ src,
                                const int* __restrict__ dst,
                                float* __restrict__ agg, int nE) {
    int t = blockIdx.x * blockDim.x + threadIdx.x;
    int e = t >> 4;
    if (e >= nE) return;
    int c4 = (t & 15) * 4;
    int s = src[e];
    int d = dst[e];
    const float4 v = *reinterpret_cast<const float4*>(x + (size_t)s * HID + c4);
    float* p = agg + (size_t)d * HID + c4;
    atomicAdd(p + 0, v.x);
    atomicAdd(p + 1, v.y);
    atomicAdd(p + 2, v.z);
    atomicAdd(p + 3, v.w);
}

// ---------------------------------------------------------------------------
// Fused GEMM(64x64) + bias + ReLU using V_WMMA_F32_16X16X4_F32.
//   If agg != nullptr:  A = (1+eps[layer])*X + agg  (VALU staging, bounds-safe,
//                       since X may be the harness-owned input allocation)
//   If agg == nullptr:  X is workspace-backed -> A tile staged with async DMA
//   If stats != nullptr: also accumulate per-channel sum / sum-of-squares
//                        (BatchNorm training statistics) via block reduction.
// W is staged TRANSPOSED into LDS via per-lane async b32 DMA so that both A and
// B fragments are single contiguous ds_load_b64's (no packing movs).
// Full tiles emit results via GLOBAL_STORE_ASYNC_FROM_LDS_B128.
// Block: 128 threads = 4 wave32; each wave owns a 16-row strip of a 64-row tile.
// ---------------------------------------------------------------------------
__global__ __launch_bounds__(128)
void gin_mlp_wmma(const float* __restrict__ X,
                  const float* __restrict__ agg,
                  const float* __restrict__ epsArr, int layer,
                  const float* __restrict__ W,
                  const float* __restrict__ bias,
                  float* __restrict__ Out,
                  float* __restrict__ stats, int nN) {
    __shared__ float lA[64][68];    // A tile / result tile (padded rows)
    __shared__ float lWt[64][68];   // W transposed: lWt[col][k] = W[k][col]

    const int tid  = threadIdx.x;
    const int wave = tid >> 5;       // 0..3
    const int lane = tid & 31;
    const int half = lane >> 4;      // 0: K+{0,1}, 1: K+{2,3}
    const int l16  = lane & 15;
    const int row0 = blockIdx.x * 64;
    const bool full = (row0 + 64) <= nN;   // uniform per block

    // W tile: transposing async DMA (contiguous global reads, per-lane LDS
    // scatter). Issued first so it overlaps the A-tile staging below.
    for (int f = tid; f < 64 * 64; f += 128) {
        int k = f >> 6, c = f & 63;
        gin_async_ld_b32((unsigned)(size_t)&lWt[c][k], W + f);
    }

    if (agg) {
        // GINConv combine, bounds-checked VALU staging
        const float epsv = 1.0f + epsArr[layer];
        for (int i = tid; i < 64 * 64; i += 128) {
            int r = i >> 6, c = i & 63;
            int g = row0 + r;
            float v = 0.0f;
            if (g < nN)
                v = epsv * X[(size_t)g * HID + c] + agg[(size_t)g * HID + c];
            lA[r][c] = v;
        }
    } else {
        // Raw async DMA of the A tile (X is workspace-backed; tail over-read of
        // the last block lands in the adjacent workspace region).
        for (int j = tid; j < 1024; j += 128) {
            int f = j << 2;
            int r = f >> 6, c = f & 63;
            gin_async_ld_b128((unsigned)(size_t)&lA[r][c],
                              X + (size_t)row0 * HID + f);
        }
    }
    gin_async_wait0();
    __syncthreads();

    const int mbase = wave * 16;

    v8f acc[4];
#pragma unroll
    for (int t = 0; t < 4; ++t)
#pragma unroll
        for (int i = 0; i < 8; ++i) acc[t][i] = 0.0f;

    const int arow = mbase + l16;
    const int koff = 2 * half;
#pragma unroll
    for (int k0 = 0; k0 < 64; k0 += 4) {
        // A fragment: 16x4 f32; lanes 0-15 carry K={k0,k0+1}, lanes 16-31 K={k0+2,k0+3}
        v2f a = *reinterpret_cast<const v2f*>(&lA[arow][k0 + koff]);
#pragma unroll
        for (int t = 0; t < 4; ++t) {
            int col = t * 16 + l16;
            // B fragment: 4x16 f32, mirrored half/K striping; contiguous in lWt
            v2f b = *reinterpret_cast<const v2f*>(&lWt[col][k0 + koff]);
            acc[t] = __builtin_amdgcn_wmma_f32_16x16x4_f32(
                false, a, false, b, (short)0, acc[t], false, false);
        }
    }
    __syncthreads();   // all waves done reading lA before it becomes the result tile

    // Epilogue: bias + ReLU into LDS; C/D layout VGPR i -> M = i + 8*half, N = l16
#pragma unroll
    for (int t = 0; t < 4; ++t) {
        int col = t * 16 + l16;
        float bv = bias[col];
#pragma unroll
        for (int i = 0; i < 8; ++i) {
            int r = mbase + half * 8 + i;
            int g = row0 + r;
            float v = acc[t][i] + bv;
            v = v > 0.0f ? v : 0.0f;
            lA[r][col] = (g < nN) ? v : 0.0f;       // masked result tile
            if (!full && g < nN) Out[(size_t)g * HID + col] = v;
        }
    }
    __syncthreads();

    if (full) {
        // Emit the whole 64x64 result tile via async LDS->global B128 stores.
        for (int j = tid; j < 1024; j += 128) {
            int f = j << 2;
            int r = f >> 6, c = f & 63;
            gin_async_st_b128(Out + (size_t)row0 * HID + f,
                              (unsigned)(size_t)&lA[r][c]);
        }
    }

    if (stats) {  // uniform branch: per-channel partial sums, 128 atomics/block
        if (tid < 64) {
            float s = 0.0f, s2 = 0.0f;
#pragma unroll 8
            for (int r = 0; r < 64; ++r) {
                float v = lA[r][tid];
                s += v;
                s2 += v * v;
            }
            atomicAdd(&stats[tid], s);
            atomicAdd(&stats[64 + tid], s2);
        }
    }
    if (full) gin_async_wait0();
}

// ---------------------------------------------------------------------------
// BatchNorm: finalize stats -> mean, invstd; then apply in place (float4).
// ---------------------------------------------------------------------------
__global__ void gin_bn_finalize(float* __restrict__ stats, float n) {
    int c = threadIdx.x;  // 64 threads
    float mean = stats[c] / n;
    float var  = stats[64 + c] / n - mean * mean;
    stats[128 + c] = mean;
    stats[192 + c] = rsqrtf(var + 1e-5f);
}

__global__ void gin_bn_apply(float* __restrict__ h,
                             const float* __restrict__ gamma,
                             const float* __restrict__ beta,
                             const float* __restrict__ stats, int nq) {
    int i = blockIdx.x * blockDim.x + threadIdx.x;  // float4 index
    if (i >= nq) return;
    int c4 = (i & 15) * 4;                          // channel group
    float4 v = reinterpret_cast<float4*>(h)[i];
    v.x = gamma[c4 + 0] * (v.x - stats[128 + c4 + 0]) * stats[192 + c4 + 0] + beta[c4 + 0];
    v.y = gamma[c4 + 1] * (v.y - stats[128 + c4 + 1]) * stats[192 + c4 + 1] + beta[c4 + 1];
    v.z = gamma[c4 + 2] * (v.z - stats[128 + c4 + 2]) * stats[192 + c4 + 2] + beta[c4 + 2];
    v.w = gamma[c4 + 3] * (v.w - stats[128 + c4 + 3]) * stats[192 + c4 + 3] + beta[c4 + 3];
    reinterpret_cast<float4*>(h)[i] = v;
}

// ---------------------------------------------------------------------------
// Global mean pool by graph id (batch), via atomics + divide.
// ---------------------------------------------------------------------------
__global__ void gin_pool_accum(const float* __restrict__ x,
                               const int* __restrict__ batch,
                               float* __restrict__ out,
                               float* __restrict__ counts, int nN) {
    int i = blockIdx.x * blockDim.x + threadIdx.x;
    if (i >= nN * HID) return;
    int node = i >> 6;
    int c = i & 63;
    int b = batch[node];
    atomicAdd(&out[(size_t)b * HID + c], x[i]);
    if (c == 0) atomicAdd(&counts[b], 1.0f);
}

__global__ void gin_pool_div(float* __restrict__ out,
                             const float* __restrict__ counts, int n) {
    int i = blockIdx.x * blockDim.x + threadIdx.x;
    if (i >= n) return;
    float cnt = counts[i >> 6];
    out[i] /= (cnt > 1.0f ? cnt : 1.0f);
}

// ---------------------------------------------------------------------------
// Launch
// ---------------------------------------------------------------------------
extern "C" void kernel_launch(void* const* d_in, const int* in_sizes, int n_in,
                              void* d_out, int out_size, void* d_ws, size_t ws_size,
                              hipStream_t stream) {
    const float* x     = (const float*)d_in[0];
    const int*   eidx  = (const int*)d_in[1];   // (2, E): [src | dst]
    const int*   batch = (const int*)d_in[2];
    const float* W1    = (const float*)d_in[3]; // (4,64,64)
    const float* b1    = (const float*)d_in[4]; // (4,64)
    const float* W2    = (const float*)d_in[5];
    const float* b2    = (const float*)d_in[6];
    const float* gamma = (const float*)d_in[7];
    const float* beta  = (const float*)d_in[8];
    const float* eps   = (const float*)d_in[9]; // (4,)

    const int nN = in_sizes[0] / HID;
    const int nE = in_sizes[1] / 2;

    float* P0     = (float*)d_ws;                  // ping buffer (x / t2)
    float* P1     = P0 + (size_t)nN * HID;         // pong buffer (t1)
    float* agg    = P1 + (size_t)nN * HID;         // neighbor aggregation
    float* stats  = agg + (size_t)nN * HID;        // 256 floats: sum|sumsq|mean|invstd
    float* counts = stats + 256;                   // 512 floats
    float* out    = (float*)d_out;                 // (512, 64)

    const int n64 = nN * HID;
    const int zgrid  = (n64 + 255) / 256;
    const int sgrid  = ((nE * 16) + 255) / 256;
    const int mgrid  = (nN + 63) / 64;
    const int qgrid  = ((n64 / 4) + 255) / 256;

    const float* xcur = x;
    for (int l = 0; l < LAYERS; ++l) {
        gin_zero_f32<<<zgrid, 256, 0, stream>>>(agg, n64);
        gin_zero_f32<<<1, 256, 0, stream>>>(stats, 256);
        gin_scatter_add<<<sgrid, 256, 0, stream>>>(xcur, eidx, eidx + nE, agg, nE);
        // t1 = relu(((1+eps)x + agg) @ W1 + b1)
        gin_mlp_wmma<<<mgrid, 128, 0, stream>>>(
            xcur, agg, eps, l, W1 + (size_t)l * HID * HID, b1 + (size_t)l * HID,
            P1, nullptr, nN);
        // t2 = relu(t1 @ W2 + b2), with BN statistics accumulation
        gin_mlp_wmma<<<mgrid, 128, 0, stream>>>(
            P1, nullptr, eps, l, W2 + (size_t)l * HID * HID, b2 + (size_t)l * HID,
            P0, stats, nN);
        gin_bn_finalize<<<1, 64, 0, stream>>>(stats, (float)nN);
        gin_bn_apply<<<qgrid, 256, 0, stream>>>(P0, gamma + (size_t)l * HID,
                                                beta + (size_t)l * HID, stats, n64 / 4);
        xcur = P0;
    }

    gin_zero_f32<<<(out_size + 255) / 256, 256, 0, stream>>>(out, out_size);
    gin_zero_f32<<<(512 + 255) / 256, 256, 0, stream>>>(counts, 512);
    gin_pool_accum<<<zgrid, 256, 0, stream>>>(xcur, batch, out, counts, nN);
    gin_pool_div<<<(out_size + 255) / 256, 256, 0, stream>>>(out, counts, out_size);
}